// GASConv_41377714929861
// MI455X (gfx1250) — compile-verified
//
#include <hip/hip_runtime.h>
#include <hip/hip_bf16.h>
#include <math.h>

typedef __attribute__((ext_vector_type(2))) float v2f;
typedef __attribute__((ext_vector_type(8))) float v8f;

// ---------------------------------------------------------------------------
// WMMA helpers: V_WMMA_F32_16X16X4_F32, wave32.
// A (16x4 f32): lane l holds rows m=l&15; VGPR0/1 = K = (l>>4)*2 + {0,1}
// B (4x16 f32): lane l holds col n=l&15; VGPR v = K = (l>>4)*2 + v
// C/D (16x16 f32): VGPR r, lane l -> M = r + (l>>4)*8, N = l&15
// ---------------------------------------------------------------------------
template <int NK>
__device__ __forceinline__ v8f wmma_acc(const v2f* __restrict__ afr,
                                        const float* __restrict__ W, int ldb,
                                        int ncol0, float cinit, int lane) {
    v8f c;
#pragma unroll
    for (int r = 0; r < 8; ++r) c[r] = cinit;
    const int n   = ncol0 + (lane & 15);
    const int khB = (lane >> 4) << 1;
#pragma unroll
    for (int kk = 0; kk < NK; ++kk) {
        v2f b;
        b.x = W[(size_t)(kk * 4 + khB + 0) * ldb + n];
        b.y = W[(size_t)(kk * 4 + khB + 1) * ldb + n];
        c = __builtin_amdgcn_wmma_f32_16x16x4_f32(
                /*neg_a=*/false, afr[kk], /*neg_b=*/false, b,
                /*c_mod=*/(short)0, c, /*reuse_a=*/false, /*reuse_b=*/false);
    }
    return c;
}

__device__ __forceinline__ void store_tile(v8f c, float* __restrict__ out,
                                           int ldo, int row0, int ncol0, int lane) {
    const int n     = ncol0 + (lane & 15);
    const int rbase = row0 + ((lane >> 4) << 3);
#pragma unroll
    for (int r = 0; r < 8; ++r) out[(size_t)(rbase + r) * ldo + n] = c[r];
}

// ---------------------------------------------------------------------------
// Node-side fused transform: he = X@W1+b1 (64), att = X@W2+b2 (128),
// hout[:,0:32] = X@W3+b3 (written with row-stride 64 into the final output).
// One wave per 16-row tile; A fragments reused across all 14 N-tiles.
// ---------------------------------------------------------------------------
__global__ void __launch_bounds__(32)
node_fused_kernel(const float* __restrict__ X,
                  const float* __restrict__ W1, const float* __restrict__ b1,
                  const float* __restrict__ W2, const float* __restrict__ b2,
                  const float* __restrict__ W3, const float* __restrict__ b3,
                  float* __restrict__ he, float* __restrict__ att,
                  float* __restrict__ hout) {
    const int lane = threadIdx.x;
    const int row0 = blockIdx.x << 4;
    const int m    = lane & 15;
    const int khA  = (lane >> 4) << 1;
    const float* xr = X + (size_t)(row0 + m) * 64 + khA;
    v2f afr[16];
#pragma unroll
    for (int kk = 0; kk < 16; ++kk) afr[kk] = *(const v2f*)(xr + kk * 4);
    const int nl = lane & 15;
    for (int t = 0; t < 4; ++t) {
        v8f c = wmma_acc<16>(afr, W1, 64, t * 16, b1[t * 16 + nl], lane);
        store_tile(c, he, 64, row0, t * 16, lane);
    }
    for (int t = 0; t < 8; ++t) {
        v8f c = wmma_acc<16>(afr, W2, 128, t * 16, b2[t * 16 + nl], lane);
        store_tile(c, att, 128, row0, t * 16, lane);
    }
    for (int t = 0; t < 2; ++t) {
        v8f c = wmma_acc<16>(afr, W3, 32, t * 16, b3[t * 16 + nl], lane);
        store_tile(c, hout, 64, row0, t * 16, lane);
    }
}

// ---------------------------------------------------------------------------
// Edge GEMM: out = E@We + be + he_s[src] + he_d[dst]   (gathers fused in epilogue)
// ---------------------------------------------------------------------------
__global__ void __launch_bounds__(32)
edge_gemm_kernel(const float* __restrict__ Ef,
                 const float* __restrict__ We, const float* __restrict__ be,
                 const float* __restrict__ he_s, const int* __restrict__ src,
                 const float* __restrict__ he_d, const int* __restrict__ dst,
                 float* __restrict__ out) {
    const int lane = threadIdx.x;
    const int row0 = blockIdx.x << 4;
    const int m    = lane & 15;
    const int khA  = (lane >> 4) << 1;
    const float* xr = Ef + (size_t)(row0 + m) * 64 + khA;
    v2f afr[16];
#pragma unroll
    for (int kk = 0; kk < 16; ++kk) afr[kk] = *(const v2f*)(xr + kk * 4);
    const int nl    = lane & 15;
    const int rbase = row0 + ((lane >> 4) << 3);
    for (int t = 0; t < 4; ++t) {
        v8f c = wmma_acc<16>(afr, We, 64, t * 16, be[t * 16 + nl], lane);
        const int n = t * 16 + nl;
#pragma unroll
        for (int r = 0; r < 8; ++r) {
            const int row = rbase + r;
            const int s = src[row], d = dst[row];
            out[(size_t)row * 64 + n] =
                c[r] + he_s[(size_t)s * 64 + n] + he_d[(size_t)d * 64 + n];
        }
    }
}

// ---------------------------------------------------------------------------
// Attention score + segment max  (wave per edge; 128-dim dot, shfl reduce)
// ---------------------------------------------------------------------------
__device__ __forceinline__ void atomicMaxF(float* addr, float v) {
    if (v >= 0.0f) atomicMax((int*)addr, __float_as_int(v));
    else           atomicMin((unsigned int*)addr, __float_as_uint(v));
}

__global__ void edge_score_kernel(const float* __restrict__ nsrc,
                                  const float* __restrict__ ef,
                                  const int* __restrict__ src,
                                  const int* __restrict__ dst,
                                  const float* __restrict__ attd,
                                  float* __restrict__ s_out,
                                  float* __restrict__ mx, int E) {
    const int lane = threadIdx.x & 31;
    int wid = (int)((blockIdx.x * blockDim.x + threadIdx.x) >> 5);
    const int nw = (int)((gridDim.x * blockDim.x) >> 5);
    for (int e = wid; e < E; e += nw) {
        const int se = src[e], de = dst[e];
        const float* hp = (lane < 16)
            ? (nsrc + (size_t)se * 64 + lane * 4)
            : (ef + (size_t)e * 64 + (lane - 16) * 4);
        const float4 h = *(const float4*)hp;
        const float4 a = *(const float4*)(attd + (size_t)de * 128 + lane * 4);
        float p = h.x * a.x + h.y * a.y + h.z * a.z + h.w * a.w;
#pragma unroll
        for (int off = 16; off > 0; off >>= 1) p += __shfl_xor(p, off, 32);
        if (lane == 0) {
            s_out[e] = p;
            atomicMaxF(mx + de, p);
        }
    }
}

// exp(s - max[dst]) + segment sum (thread per edge; exp stored over s)
__global__ void edge_exp_kernel(float* __restrict__ s, const int* __restrict__ dst,
                                const float* __restrict__ mx, float* __restrict__ sm,
                                int E) {
    const int e = blockIdx.x * blockDim.x + threadIdx.x;
    if (e >= E) return;
    const int de = dst[e];
    const float ex = __expf(s[e] - mx[de]);
    s[e] = ex;
    atomicAdd(sm + de, ex);
}

// weighted scatter-add of concat(node[src], edge) into agg[dst]  (wave per edge)
__global__ void edge_agg_kernel(const float* __restrict__ nsrc,
                                const float* __restrict__ ef,
                                const int* __restrict__ src,
                                const int* __restrict__ dst,
                                const float* __restrict__ exs,
                                const float* __restrict__ sm,
                                float* __restrict__ agg, int E) {
    const int lane = threadIdx.x & 31;
    int wid = (int)((blockIdx.x * blockDim.x + threadIdx.x) >> 5);
    const int nw = (int)((gridDim.x * blockDim.x) >> 5);
    for (int e = wid; e < E; e += nw) {
        const int se = src[e], de = dst[e];
        const float w = exs[e] / sm[de];
        const float* hp = (lane < 16)
            ? (nsrc + (size_t)se * 64 + lane * 4)
            : (ef + (size_t)e * 64 + (lane - 16) * 4);
        const float4 h = *(const float4*)hp;
        float* ap = agg + (size_t)de * 128 + lane * 4;
        atomicAdd(ap + 0, h.x * w);
        atomicAdd(ap + 1, h.y * w);
        atomicAdd(ap + 2, h.z * w);
        atomicAdd(ap + 3, h.w * w);
    }
}

// ---------------------------------------------------------------------------
// Output GEMM: [N,128] @ [128,32] + b  -> output columns 32..63 (ld = 64)
// ---------------------------------------------------------------------------
__global__ void __launch_bounds__(32)
agg_gemm_kernel(const float* __restrict__ A, const float* __restrict__ W,
                const float* __restrict__ b, float* __restrict__ out) {
    const int lane = threadIdx.x;
    const int row0 = blockIdx.x << 4;
    const int m    = lane & 15;
    const int khA  = (lane >> 4) << 1;
    const float* xr = A + (size_t)(row0 + m) * 128 + khA;
    v2f afr[32];
#pragma unroll
    for (int kk = 0; kk < 32; ++kk) afr[kk] = *(const v2f*)(xr + kk * 4);
    const int nl = lane & 15;
    for (int t = 0; t < 2; ++t) {
        v8f c = wmma_acc<32>(afr, W, 32, t * 16, b[t * 16 + nl], lane);
        store_tile(c, out, 64, row0, t * 16, lane);
    }
}

__global__ void fill_kernel(unsigned* __restrict__ p, unsigned v, size_t n) {
    size_t i  = (size_t)blockIdx.x * blockDim.x + threadIdx.x;
    size_t st = (size_t)gridDim.x * blockDim.x;
    for (; i < n; i += st) p[i] = v;
}

// ---------------------------------------------------------------------------
extern "C" void kernel_launch(void* const* d_in, const int* in_sizes, int n_in,
                              void* d_out, int out_size, void* d_ws, size_t ws_size,
                              hipStream_t stream) {
    (void)n_in; (void)out_size; (void)ws_size;
    const float* f_feat = (const float*)d_in[0];
    const float* b_feat = (const float*)d_in[1];
    const float* u_feat = (const float*)d_in[2];
    const float* v_feat = (const float*)d_in[3];
    const int*   fsrc   = (const int*)d_in[4];
    const int*   fdst   = (const int*)d_in[5];
    const int*   bsrc   = (const int*)d_in[6];
    const int*   bdst   = (const int*)d_in[7];
    const float* We  = (const float*)d_in[8];   const float* be  = (const float*)d_in[9];
    const float* Wul = (const float*)d_in[10];  const float* bul = (const float*)d_in[11];
    const float* Wvl = (const float*)d_in[12];  const float* bvl = (const float*)d_in[13];
    const float* Wau = (const float*)d_in[14];  const float* bau = (const float*)d_in[15];
    const float* Wav = (const float*)d_in[16];  const float* bav = (const float*)d_in[17];
    const float* Wwu = (const float*)d_in[18];  const float* bwu = (const float*)d_in[19];
    const float* Wwv = (const float*)d_in[20];  const float* bwv = (const float*)d_in[21];
    const float* WVu = (const float*)d_in[22];  const float* bVu = (const float*)d_in[23];
    const float* WVv = (const float*)d_in[24];  const float* bVv = (const float*)d_in[25];

    const int NU = in_sizes[2] / 64;
    const int NV = in_sizes[3] / 64;
    const int EF = in_sizes[4];
    const int EB = in_sizes[6];

    // output layout: hf [EF,64] | hb [EB,64] | hu [NU,64] | hv [NV,64]
    float* hf = (float*)d_out;
    float* hb = hf + (size_t)EF * 64;
    float* hu = hb + (size_t)EB * 64;
    float* hv = hu + (size_t)NU * 64;

    // workspace layout
    float* ws = (float*)d_ws;
    size_t off = 0;
    float* he_u  = ws + off; off += (size_t)NU * 64;
    float* he_v  = ws + off; off += (size_t)NV * 64;
    float* att_u = ws + off; off += (size_t)NU * 128;
    float* att_v = ws + off; off += (size_t)NV * 128;
    float* agg_u = ws + off; off += (size_t)NU * 128;
    float* agg_v = ws + off; off += (size_t)NV * 128;
    float* s_f   = ws + off; off += (size_t)EF;
    float* s_b   = ws + off; off += (size_t)EB;
    float* mx_u  = ws + off; off += (size_t)NU;
    float* mx_v  = ws + off; off += (size_t)NV;
    float* sm_u  = ws + off; off += (size_t)NU;
    float* sm_v  = ws + off; off += (size_t)NV;

    // re-init scratch every call (graph replay must be deterministic)
    const unsigned NEG_INF = 0xFF800000u;
    fill_kernel<<<2048, 256, 0, stream>>>((unsigned*)agg_u, 0u, (size_t)NU * 128);
    fill_kernel<<<2048, 256, 0, stream>>>((unsigned*)agg_v, 0u, (size_t)NV * 128);
    fill_kernel<<<256, 256, 0, stream>>>((unsigned*)sm_u, 0u, (size_t)NU);
    fill_kernel<<<256, 256, 0, stream>>>((unsigned*)sm_v, 0u, (size_t)NV);
    fill_kernel<<<256, 256, 0, stream>>>((unsigned*)mx_u, NEG_INF, (size_t)NU);
    fill_kernel<<<256, 256, 0, stream>>>((unsigned*)mx_v, NEG_INF, (size_t)NV);

    // node-side fused transforms (he, att, and output cols 0..31)
    node_fused_kernel<<<NU / 16, 32, 0, stream>>>(u_feat, Wul, bul, Wau, bau,
                                                  WVu, bVu, he_u, att_u, hu);
    node_fused_kernel<<<NV / 16, 32, 0, stream>>>(v_feat, Wvl, bvl, Wav, bav,
                                                  WVv, bVv, he_v, att_v, hv);

    // edge GEMMs with fused gather-adds
    edge_gemm_kernel<<<EF / 16, 32, 0, stream>>>(f_feat, We, be, he_u, fsrc,
                                                 he_v, fdst, hf);
    edge_gemm_kernel<<<EB / 16, 32, 0, stream>>>(b_feat, We, be, he_v, bsrc,
                                                 he_u, bdst, hb);

    // edge softmax (3 passes) + weighted aggregation
    edge_score_kernel<<<2048, 256, 0, stream>>>(u_feat, f_feat, fsrc, fdst,
                                                att_v, s_f, mx_v, EF);
    edge_score_kernel<<<2048, 256, 0, stream>>>(v_feat, b_feat, bsrc, bdst,
                                                att_u, s_b, mx_u, EB);
    edge_exp_kernel<<<(EF + 255) / 256, 256, 0, stream>>>(s_f, fdst, mx_v, sm_v, EF);
    edge_exp_kernel<<<(EB + 255) / 256, 256, 0, stream>>>(s_b, bdst, mx_u, sm_u, EB);
    edge_agg_kernel<<<2048, 256, 0, stream>>>(u_feat, f_feat, fsrc, fdst,
                                              s_f, sm_v, agg_v, EF);
    edge_agg_kernel<<<2048, 256, 0, stream>>>(v_feat, b_feat, bsrc, bdst,
                                              s_b, sm_u, agg_u, EB);

    // output GEMMs into columns 32..63
    agg_gemm_kernel<<<NU / 16, 32, 0, stream>>>(agg_u, Wwu, bwu, hu + 32);
    agg_gemm_kernel<<<NV / 16, 32, 0, stream>>>(agg_v, Wwv, bwv, hv + 32);
}